// DHSA_51384988729455
// MI455X (gfx1250) — compile-verified
//
#include <hip/hip_runtime.h>

// Problem constants (match setup_inputs)
#define BB   2
#define CC   64
#define HH   192
#define WW   192
#define HWN  36864      // H*W
#define C5   320        // 5*c
#define C15  960        // 15*c
#define NHEADS 4
#define DDIM 64         // ch*factor = 16*4
#define NNN  9216       // hw/factor
#define SORTN 65536     // pow2 pad for 36864-element row sorts
#define NROWS 128       // B*C rows for v-sort

typedef __attribute__((ext_vector_type(2))) float v2f;
typedef __attribute__((ext_vector_type(8))) float v8f;

// CDNA5 fp32 WMMA: D = A(16x4) x B(4x16) + C(16x16), wave32, one matrix per wave.
__device__ __forceinline__ v8f wmma4(v2f a, v2f b, v8f c) {
  return __builtin_amdgcn_wmma_f32_16x16x4_f32(false, a, false, b, (short)0, c, false, false);
}

// ---------------------------------------------------------------------------
// 256-slot LDS bitonic argsort of a 192-long column along H (first 32 ch)
// ---------------------------------------------------------------------------
__global__ __launch_bounds__(256) void k_sort_h(const float* __restrict__ x,
                                                float* __restrict__ xcat,
                                                int* __restrict__ idx_h) {
  int col = blockIdx.x;                 // over B*32*W
  int j  = col % WW;
  int ci = (col / WW) % 32;
  int bb = col / (WW * 32);
  int t = threadIdx.x;
  __shared__ float sk[256];
  __shared__ int   si[256];
  float v = __builtin_inff();
  int   id = 0x7fffffff;
  if (t < HH) { v = x[(((size_t)bb * CC + ci) * HH + t) * WW + j]; id = t; }
  sk[t] = v; si[t] = id;
  __syncthreads();
  for (int k = 2; k <= 256; k <<= 1)
    for (int jj = k >> 1; jj > 0; jj >>= 1) {
      int l = t ^ jj;
      if (l > t) {
        bool up = ((t & k) == 0);
        float a = sk[t], b = sk[l];
        if ((a > b) == up) {
          sk[t] = b; sk[l] = a;
          int tmp = si[t]; si[t] = si[l]; si[l] = tmp;
        }
      }
      __syncthreads();
    }
  if (t < HH) {
    idx_h[(((size_t)bb * 32 + ci) * HH + t) * WW + j] = si[t];
    xcat[(((size_t)bb * CC + ci) * HH + t) * WW + j]  = sk[t];
  }
}

// Sort each row (along W) of xcat's first 32 channels, in place; emit idx_w.
__global__ __launch_bounds__(256) void k_sort_w(float* __restrict__ xcat,
                                                int* __restrict__ idx_w) {
  int row = blockIdx.x;                 // over B*32*H
  int i  = row % HH;
  int ci = (row / HH) % 32;
  int bb = row / (HH * 32);
  int t = threadIdx.x;
  __shared__ float sk[256];
  __shared__ int   si[256];
  float* base = xcat + (((size_t)bb * CC + ci) * HH + i) * WW;
  float v = __builtin_inff();
  int   id = 0x7fffffff;
  if (t < WW) { v = base[t]; id = t; }
  sk[t] = v; si[t] = id;
  __syncthreads();
  for (int k = 2; k <= 256; k <<= 1)
    for (int jj = k >> 1; jj > 0; jj >>= 1) {
      int l = t ^ jj;
      if (l > t) {
        bool up = ((t & k) == 0);
        float a = sk[t], b = sk[l];
        if ((a > b) == up) {
          sk[t] = b; sk[l] = a;
          int tmp = si[t]; si[t] = si[l]; si[l] = tmp;
        }
      }
      __syncthreads();
    }
  if (t < WW) {
    idx_w[(((size_t)bb * 32 + ci) * HH + i) * WW + t] = si[t];
    base[t] = sk[t];
  }
}

__global__ void k_copy_hi(const float* __restrict__ x, float* __restrict__ xcat) {
  long long idx = (long long)blockIdx.x * blockDim.x + threadIdx.x;
  if (idx >= (long long)BB * 32 * HWN) return;
  int m  = idx % HWN;
  int ch = (idx / HWN) % 32;
  int bb = idx / ((long long)HWN * 32);
  size_t o = ((size_t)bb * CC + 32 + ch) * HWN + m;
  xcat[o] = x[o];
}

// ---------------------------------------------------------------------------
// WMMA fp32 GEMM: Y[M,N] = A[M,K] * B[K,N] (+bias[M]), batched over blockIdx.z
// One wave computes one 16(M) x 64(N) strip: 4 accumulators sharing the A
// fragment (4x A reuse per load, 4 WMMAs issued per K-step). N % 64 == 0.
// ---------------------------------------------------------------------------
__global__ __launch_bounds__(256) void k_gemm_wn(
    const float* __restrict__ A, long long aBatch,
    const float* __restrict__ Bm, long long bBatch,
    const float* __restrict__ bias,
    float* __restrict__ Y, long long yBatch,
    int M, int K, int N) {
  int wave = threadIdx.x >> 5;
  int lane = threadIdx.x & 31;
  int g    = lane >> 4;                 // half-wave group
  int sel  = lane & 15;
  int ntiles = N >> 4, mtiles = M >> 4;
  int ngrp = ntiles >> 2;               // groups of 4 consecutive N tiles
  int tile = blockIdx.x * 8 + wave;     // wave-uniform -> EXEC stays all-ones
  if (tile >= mtiles * ngrp) return;
  int mt = tile / ngrp;
  int ng = tile - mt * ngrp;
  const float* Ab = A  + (long long)blockIdx.z * aBatch;
  const float* Bb = Bm + (long long)blockIdx.z * bBatch;
  float*       Yb = Y  + (long long)blockIdx.z * yBatch;
  int arow  = mt * 16 + sel;
  int nbase = ng * 64 + sel;
  v8f acc0 = {0.f,0.f,0.f,0.f,0.f,0.f,0.f,0.f};
  v8f acc1 = acc0, acc2 = acc0, acc3 = acc0;
  const float* ap = Ab + (size_t)arow * K + 2 * g;
  const float* bp = Bb + (size_t)(2 * g) * N + nbase;
  for (int k = 0; k < K; k += 4) {
    v2f a;
    a.x = ap[0]; a.y = ap[1];
    v2f b0, b1, b2, b3;
    b0.x = bp[0];  b0.y = bp[(size_t)N];
    b1.x = bp[16]; b1.y = bp[(size_t)N + 16];
    b2.x = bp[32]; b2.y = bp[(size_t)N + 32];
    b3.x = bp[48]; b3.y = bp[(size_t)N + 48];
    acc0 = wmma4(a, b0, acc0);
    acc1 = wmma4(a, b1, acc1);
    acc2 = wmma4(a, b2, acc2);
    acc3 = wmma4(a, b3, acc3);
    ap += 4;
    bp += (size_t)4 * N;
  }
#pragma unroll
  for (int v = 0; v < 8; ++v) {
    int row = mt * 16 + v + 8 * g;
    float bv = bias ? bias[row] : 0.f;
    float* yr = Yb + (size_t)row * N + nbase;
    yr[0]  = acc0[v] + bv;
    yr[16] = acc1[v] + bv;
    yr[32] = acc2[v] + bv;
    yr[48] = acc3[v] + bv;
  }
}

// WMMA fp32 GEMM, B transposed: Y[M,N] = A[M,K] * Bt[N,K]^T  (S = Q * K^T)
__global__ __launch_bounds__(256) void k_gemm_nt(
    const float* __restrict__ A, long long aBatch,
    const float* __restrict__ Bt, long long bBatch,
    float* __restrict__ Y, long long yBatch,
    int M, int K, int N) {
  int wave = threadIdx.x >> 5;
  int lane = threadIdx.x & 31;
  int g    = lane >> 4;
  int sel  = lane & 15;
  int ntiles = N >> 4, mtiles = M >> 4;
  int tile = blockIdx.x * 8 + wave;
  if (tile >= mtiles * ntiles) return;
  int mt = tile / ntiles;
  int nt = tile - mt * ntiles;
  const float* Ab = A  + (long long)blockIdx.z * aBatch;
  const float* Bb = Bt + (long long)blockIdx.z * bBatch;
  float*       Yb = Y  + (long long)blockIdx.z * yBatch;
  int arow = mt * 16 + sel;
  int brow = nt * 16 + sel;
  v8f acc = {0.f,0.f,0.f,0.f,0.f,0.f,0.f,0.f};
  const float* ap = Ab + (size_t)arow * K + 2 * g;
  const float* bp = Bb + (size_t)brow * K + 2 * g;
  for (int k = 0; k < K; k += 4) {
    v2f a, b;
    a.x = ap[0]; a.y = ap[1];
    b.x = bp[0]; b.y = bp[1];
    acc = wmma4(a, b, acc);
    ap += 4;
    bp += 4;
  }
#pragma unroll
  for (int v = 0; v < 8; ++v) {
    int row = mt * 16 + v + 8 * g;
    Yb[(size_t)row * N + nt * 16 + sel] = acc[v];
  }
}

// ---------------------------------------------------------------------------
// Depthwise 3x3 / 5x5 / 7x7 -> cat (b, 960, h, w)
// ---------------------------------------------------------------------------
__global__ __launch_bounds__(256) void k_dwconv(
    const float* __restrict__ q,
    const float* __restrict__ w3, const float* __restrict__ w5,
    const float* __restrict__ w7, float* __restrict__ cat) {
  long long idx = (long long)blockIdx.x * blockDim.x + threadIdx.x;
  if (idx >= (long long)BB * C15 * HWN) return;
  int j  = idx % WW;
  int i  = (idx / WW) % HH;
  int oc = (idx / HWN) % C15;
  int bb = (int)(idx / ((long long)HWN * C15));
  int which = oc / C5;
  int ch = oc - which * C5;
  int ksz = 3 + 2 * which;
  int pad = 1 + which;
  const float* wp = (which == 0) ? (w3 + ch * 9)
                  : (which == 1) ? (w5 + ch * 25)
                                 : (w7 + ch * 49);
  const float* src = q + ((size_t)bb * C5 + ch) * HWN;
  float s = 0.f;
  for (int dy = 0; dy < ksz; ++dy) {
    int yy = i + dy - pad;
    if (yy < 0 || yy >= HH) continue;
    for (int dx = 0; dx < ksz; ++dx) {
      int xx = j + dx - pad;
      if (xx < 0 || xx >= WW) continue;
      s += wp[dy * ksz + dx] * src[yy * WW + xx];
    }
  }
  cat[idx] = s;
}

// ---------------------------------------------------------------------------
// v argsort: init padded key/idx rows, then single-workgroup global bitonic
// ---------------------------------------------------------------------------
__global__ void k_initsort(const float* __restrict__ qkv,
                           float* __restrict__ keys, int* __restrict__ vidx) {
  long long idx = (long long)blockIdx.x * blockDim.x + threadIdx.x;
  if (idx >= (long long)NROWS * SORTN) return;
  int m = idx % SORTN;
  int r = (int)(idx / SORTN);
  int bb = r >> 6, ch = r & 63;
  float v = __builtin_inff();
  int id = 0x7fffffff;
  if (m < HWN) {
    v = qkv[((size_t)bb * C5 + 4 * CC + ch) * HWN + m];  // v = channels [256,320)
    id = m;
  }
  keys[idx] = v; vidx[idx] = id;
}

__global__ __launch_bounds__(1024) void k_bitonic(float* __restrict__ keys,
                                                  int* __restrict__ vidx) {
  int r = blockIdx.x;
  float* kk = keys + (size_t)r * SORTN;
  int*   ii = vidx + (size_t)r * SORTN;
  for (int k = 2; k <= SORTN; k <<= 1) {
    for (int j = k >> 1; j > 0; j >>= 1) {
      for (int i = threadIdx.x; i < SORTN; i += 1024) {
        int l = i ^ j;
        if (l > i) {
          bool up = ((i & k) == 0);
          float a = kk[i], b = kk[l];
          if ((a > b) == up) {
            kk[i] = b; kk[l] = a;
            int t = ii[i]; ii[i] = ii[l]; ii[l] = t;
          }
        }
      }
      __threadfence_block();
      __syncthreads();
    }
  }
}

__global__ void k_gather(const float* __restrict__ qkv,
                         const float* __restrict__ keys,
                         const int* __restrict__ vidx,
                         float* gq1, float* gk1, float* gq2, float* gk2,
                         float* vs, int* idxv) {
  long long idx = (long long)blockIdx.x * blockDim.x + threadIdx.x;
  if (idx >= (long long)BB * CC * HWN) return;
  int m  = idx % HWN;
  int ch = (idx / HWN) % CC;
  int bb = (int)(idx / ((long long)HWN * CC));
  int r = bb * CC + ch;
  int s = vidx[(size_t)r * SORTN + m];
  const float* base = qkv + (size_t)bb * C5 * HWN;
  gq1[idx] = base[(size_t)(0 * CC + ch) * HWN + s];
  gk1[idx] = base[(size_t)(1 * CC + ch) * HWN + s];
  gq2[idx] = base[(size_t)(2 * CC + ch) * HWN + s];
  gk2[idx] = base[(size_t)(3 * CC + ch) * HWN + s];
  vs[idx]  = keys[(size_t)r * SORTN + m];
  idxv[idx] = s;
}

// prep(): (b,c,hw) -> (b,heads,64,9216) with box / non-box element mapping
__global__ void k_prep(const float* __restrict__ src, float* __restrict__ dst, int box) {
  long long idx = (long long)blockIdx.x * blockDim.x + threadIdx.x;
  if (idx >= (long long)BB * NHEADS * DDIM * NNN) return;
  int n  = idx % NNN;
  int d  = (idx / NNN) % DDIM;
  int hh = (idx / ((long long)NNN * DDIM)) % NHEADS;
  int bb = (int)(idx / ((long long)NNN * DDIM * NHEADS));
  int ci = d >> 2, f = d & 3;
  int col = box ? (f * NNN + n) : (n * 4 + f);
  dst[idx] = src[((size_t)bb * CC + hh * 16 + ci) * HWN + col];
}

__global__ __launch_bounds__(256) void k_rownorm(const float* __restrict__ src,
                                                 float* __restrict__ out) {
  int row = blockIdx.x;                 // B*HEADS*64 rows of length 9216
  const float* p = src + (size_t)row * NNN;
  float s = 0.f;
  for (int i = threadIdx.x; i < NNN; i += 256) { float v = p[i]; s += v * v; }
  __shared__ float red[256];
  red[threadIdx.x] = s;
  __syncthreads();
  for (int o = 128; o > 0; o >>= 1) {
    if (threadIdx.x < o) red[threadIdx.x] += red[threadIdx.x + o];
    __syncthreads();
  }
  if (threadIdx.x == 0) out[row] = sqrtf(red[0]);
}

// attn = exp(S/(|q||k|) * temp) / (sum_e exp + 1), in place on S
__global__ __launch_bounds__(64) void k_softmax(float* __restrict__ S,
                                                const float* __restrict__ nq,
                                                const float* __restrict__ nk,
                                                const float* __restrict__ temp) {
  int grp = blockIdx.x;                 // b*HEADS + h
  int hh = grp & 3;
  int d  = threadIdx.x;
  float* row = S + (size_t)grp * 4096 + (size_t)d * 64;
  const float* nqg = nq + grp * 64;
  const float* nkg = nk + grp * 64;
  float qn = fmaxf(nqg[d], 1e-12f);
  float tv = temp[hh];
  float sum = 0.f;
  for (int e = 0; e < 64; ++e) {
    float s = row[e] / (qn * fmaxf(nkg[e], 1e-12f)) * tv;
    sum += expf(s);
  }
  float inv = 1.f / (sum + 1.f);
  for (int e = 0; e < 64; ++e) {
    float s = row[e] / (qn * fmaxf(nkg[e], 1e-12f)) * tv;
    row[e] = expf(s) * inv;
  }
}

// un-prep both outputs, multiply, scatter back through v-sort permutation
__global__ void k_combine(const float* __restrict__ O1, const float* __restrict__ O2,
                          const int* __restrict__ idxv, float* __restrict__ prod) {
  long long idx = (long long)blockIdx.x * blockDim.x + threadIdx.x;
  if (idx >= (long long)BB * CC * HWN) return;
  int m  = idx % HWN;
  int ch = (idx / HWN) % CC;
  int bb = (int)(idx / ((long long)HWN * CC));
  int hh = ch >> 4, ci = ch & 15;
  int f1 = m / NNN, n1 = m - f1 * NNN;                     // box
  float a = O1[((size_t)(bb * NHEADS + hh) * DDIM + ci * 4 + f1) * NNN + n1];
  int f2 = m & 3, n2 = m >> 2;                             // non-box
  float b = O2[((size_t)(bb * NHEADS + hh) * DDIM + ci * 4 + f2) * NNN + n2];
  int tgt = idxv[idx];
  prod[((size_t)bb * CC + ch) * HWN + tgt] = a * b;
}

// final: undo W then H sorts on first 32 channels (scatter), copy the rest
__global__ void k_final(const float* __restrict__ po,
                        const int* __restrict__ idx_h, const int* __restrict__ idx_w,
                        float* __restrict__ out) {
  long long idx = (long long)blockIdx.x * blockDim.x + threadIdx.x;
  if (idx >= (long long)BB * CC * HWN) return;
  int j  = idx % WW;
  int i  = (idx / WW) % HH;
  int cc = (idx / HWN) % CC;
  int bb = (int)(idx / ((long long)HWN * CC));
  float v = po[idx];
  if (cc >= 32) { out[idx] = v; return; }
  int jw = idx_w[(((size_t)bb * 32 + cc) * HH + i) * WW + j];
  int io = idx_h[(((size_t)bb * 32 + cc) * HH + i) * WW + jw];
  out[(((size_t)bb * CC + cc) * HH + io) * WW + jw] = v;
}

// ---------------------------------------------------------------------------
extern "C" void kernel_launch(void* const* d_in, const int* in_sizes, int n_in,
                              void* d_out, int out_size, void* d_ws, size_t ws_size,
                              hipStream_t stream) {
  (void)in_sizes; (void)n_in; (void)out_size; (void)ws_size;
  const float* x      = (const float*)d_in[0];
  const float* temp   = (const float*)d_in[1];
  const float* w_qkv  = (const float*)d_in[2];
  const float* w3     = (const float*)d_in[3];
  const float* w5     = (const float*)d_in[4];
  const float* w7     = (const float*)d_in[5];
  const float* w_fuse = (const float*)d_in[6];
  const float* b_fuse = (const float*)d_in[7];
  const float* w_proj = (const float*)d_in[8];
  float* out = (float*)d_out;

  // workspace carve-out (~900 MB total)
  char* ws = (char*)d_ws;
  size_t cur = 0;
  auto alloc = [&](size_t bytes) -> void* {
    void* p = ws + cur;
    cur += (bytes + 255) & ~(size_t)255;
    return p;
  };
  const size_t NB = (size_t)BB * CC * HWN;        // 4,718,592 elems
  int*   idx_h = (int*)  alloc((size_t)BB * 32 * HWN * 4);
  int*   idx_w = (int*)  alloc((size_t)BB * 32 * HWN * 4);
  float* xcat  = (float*)alloc(NB * 4);
  float* qbuf  = (float*)alloc((size_t)BB * C5  * HWN * 4);
  float* catb  = (float*)alloc((size_t)BB * C15 * HWN * 4);
  float* qkvb  = (float*)alloc((size_t)BB * C5  * HWN * 4);
  float* keys  = (float*)alloc((size_t)NROWS * SORTN * 4);
  int*   vidx  = (int*)  alloc((size_t)NROWS * SORTN * 4);
  int*   idxv  = (int*)  alloc(NB * 4);
  float* gq1 = (float*)alloc(NB * 4);
  float* gk1 = (float*)alloc(NB * 4);
  float* gq2 = (float*)alloc(NB * 4);
  float* gk2 = (float*)alloc(NB * 4);
  float* vs  = (float*)alloc(NB * 4);
  float* Q1r = (float*)alloc(NB * 4);
  float* K1r = (float*)alloc(NB * 4);
  float* V1r = (float*)alloc(NB * 4);
  float* Q2r = (float*)alloc(NB * 4);
  float* K2r = (float*)alloc(NB * 4);
  float* V2r = (float*)alloc(NB * 4);
  float* nq1 = (float*)alloc(512 * 4);
  float* nk1 = (float*)alloc(512 * 4);
  float* nq2 = (float*)alloc(512 * 4);
  float* nk2 = (float*)alloc(512 * 4);
  float* S1  = (float*)alloc((size_t)8 * 4096 * 4);
  float* S2  = (float*)alloc((size_t)8 * 4096 * 4);
  float* O1  = (float*)alloc(NB * 4);
  float* O2  = (float*)alloc(NB * 4);
  float* prod= (float*)alloc(NB * 4);
  float* po  = (float*)alloc(NB * 4);

  auto cdiv = [](long long a, long long b) -> unsigned { return (unsigned)((a + b - 1) / b); };

  // 1) sort first-half channels along H, then W; copy second half
  k_sort_h<<<BB * 32 * WW, 256, 0, stream>>>(x, xcat, idx_h);
  k_sort_w<<<BB * 32 * HH, 256, 0, stream>>>(xcat, idx_w);
  k_copy_hi<<<cdiv((long long)BB * 32 * HWN, 256), 256, 0, stream>>>(x, xcat);

  // 2) qkv conv1x1: (320x64) @ (64 x 36864) per batch  [WMMA f32, 16x64 strips]
  {
    long long tiles = (C5 / 16) * (long long)(HWN / 64);
    k_gemm_wn<<<dim3(cdiv(tiles, 8), 1, BB), 256, 0, stream>>>(
        w_qkv, 0LL, xcat, (long long)CC * HWN, nullptr,
        qbuf, (long long)C5 * HWN, C5, CC, HWN);
  }

  // 3) depthwise 3/5/7 -> cat
  k_dwconv<<<cdiv((long long)BB * C15 * HWN, 256), 256, 0, stream>>>(qbuf, w3, w5, w7, catb);

  // 4) fuse conv1x1: (320x960) @ (960 x 36864) + bias  [WMMA f32, hot spot]
  {
    long long tiles = (C5 / 16) * (long long)(HWN / 64);
    k_gemm_wn<<<dim3(cdiv(tiles, 8), 1, BB), 256, 0, stream>>>(
        w_fuse, 0LL, catb, (long long)C15 * HWN, b_fuse,
        qkvb, (long long)C5 * HWN, C5, C15, HWN);
  }

  // 5) argsort v rows (36864 each), gather q1/k1/q2/k2/v
  k_initsort<<<cdiv((long long)NROWS * SORTN, 256), 256, 0, stream>>>(qkvb, keys, vidx);
  k_bitonic<<<NROWS, 1024, 0, stream>>>(keys, vidx);
  k_gather<<<cdiv((long long)NB, 256), 256, 0, stream>>>(qkvb, keys, vidx,
                                                         gq1, gk1, gq2, gk2, vs, idxv);

  // 6) prep reshapes (box and non-box variants)
  unsigned gp = cdiv((long long)NB, 256);
  k_prep<<<gp, 256, 0, stream>>>(gq1, Q1r, 1);
  k_prep<<<gp, 256, 0, stream>>>(gk1, K1r, 1);
  k_prep<<<gp, 256, 0, stream>>>(vs,  V1r, 1);
  k_prep<<<gp, 256, 0, stream>>>(gq2, Q2r, 0);
  k_prep<<<gp, 256, 0, stream>>>(gk2, K2r, 0);
  k_prep<<<gp, 256, 0, stream>>>(vs,  V2r, 0);

  // 7) L2 row norms for q/k of both attentions
  k_rownorm<<<BB * NHEADS * DDIM, 256, 0, stream>>>(Q1r, nq1);
  k_rownorm<<<BB * NHEADS * DDIM, 256, 0, stream>>>(K1r, nk1);
  k_rownorm<<<BB * NHEADS * DDIM, 256, 0, stream>>>(Q2r, nq2);
  k_rownorm<<<BB * NHEADS * DDIM, 256, 0, stream>>>(K2r, nk2);

  // 8) S = Q @ K^T per (b,head): M=64,N=64,K=9216  [WMMA f32 NT]
  {
    long long tiles = (DDIM / 16) * (long long)(DDIM / 16);  // 16
    k_gemm_nt<<<dim3(cdiv(tiles, 8), 1, BB * NHEADS), 256, 0, stream>>>(
        Q1r, (long long)DDIM * NNN, K1r, (long long)DDIM * NNN,
        S1, 4096LL, DDIM, NNN, DDIM);
    k_gemm_nt<<<dim3(cdiv(tiles, 8), 1, BB * NHEADS), 256, 0, stream>>>(
        Q2r, (long long)DDIM * NNN, K2r, (long long)DDIM * NNN,
        S2, 4096LL, DDIM, NNN, DDIM);
  }

  // 9) softmax-ish (exp / (sum + 1)) with norm + temperature scaling
  k_softmax<<<BB * NHEADS, 64, 0, stream>>>(S1, nq1, nk1, temp);
  k_softmax<<<BB * NHEADS, 64, 0, stream>>>(S2, nq2, nk2, temp);

  // 10) O = attn @ V per (b,head): M=64,K=64,N=9216  [WMMA f32, 16x64 strips]
  {
    long long tiles = (DDIM / 16) * (long long)(NNN / 64);   // 576
    k_gemm_wn<<<dim3(cdiv(tiles, 8), 1, BB * NHEADS), 256, 0, stream>>>(
        S1, 4096LL, V1r, (long long)DDIM * NNN, nullptr,
        O1, (long long)DDIM * NNN, DDIM, DDIM, NNN);
    k_gemm_wn<<<dim3(cdiv(tiles, 8), 1, BB * NHEADS), 256, 0, stream>>>(
        S2, 4096LL, V2r, (long long)DDIM * NNN, nullptr,
        O2, (long long)DDIM * NNN, DDIM, DDIM, NNN);
  }

  // 11) un-prep, multiply, scatter back through v-sort permutation
  k_combine<<<gp, 256, 0, stream>>>(O1, O2, idxv, prod);

  // 12) proj conv1x1: (64x64) @ (64 x 36864)  [WMMA f32, 16x64 strips]
  {
    long long tiles = (CC / 16) * (long long)(HWN / 64);
    k_gemm_wn<<<dim3(cdiv(tiles, 8), 1, BB), 256, 0, stream>>>(
        w_proj, 0LL, prod, (long long)CC * HWN, nullptr,
        po, (long long)CC * HWN, CC, CC, HWN);
  }

  // 13) undo H/W sorts on first half, copy second half
  k_final<<<gp, 256, 0, stream>>>(po, idx_h, idx_w, out);
}